// RAMANet_7198365188249
// MI455X (gfx1250) — compile-verified
//
#include <hip/hip_runtime.h>
#include <math.h>

// B=1024, N=16 agents, M=16 items, MENU=256, K = MENU+1 = 257 options.
// One workgroup (256 threads = 8 wave32) per batch element.
//
// paw[k,n] = sum_m allocs[k,n,m] * (w[n]*(val-bid)[n,m])
//          = GEMM:  A[257x256] (allocs rows, flat (n,m)) x Bdiag[256x16]
//            where Bdiag[j,n] = (j>>4 == n) ? vbw_flat[j] : 0
// computed with chained V_WMMA_F32_16X16X4_F32 (64 steps of K=4 per 16-row tile).

typedef __attribute__((ext_vector_type(2))) float v2f;
typedef __attribute__((ext_vector_type(8))) float v8f;

#define NAG  16
#define MENU 256
#define KOPT 257
#define TPB  256

__launch_bounds__(TPB, 2)
__global__ void auction_menu_kernel(const float* __restrict__ bids,
                                    const float* __restrict__ vals,
                                    const float* __restrict__ allocs,
                                    const float* __restrict__ wv,
                                    const float* __restrict__ bvec_g,
                                    const void*  __restrict__ temp_p,
                                    float* __restrict__ out,
                                    int B)
{
    const int b    = blockIdx.x;
    const int t    = threadIdx.x;
    const int lane = t & 31;
    const int wid  = t >> 5;
    const int half = lane >> 4;   // wave halves for WMMA fragment layout
    const int row  = lane & 15;

    // softmax_temp: harness passes a 1-element array (int 500 or float 500.0)
    float temp;
    {
        int iv = *(const int*)temp_p;
        temp = (iv >= 1 && iv < (1 << 24)) ? (float)iv : *(const float*)temp_p;
    }

    // output section offsets (return-order concatenation)
    const size_t O1 = (size_t)B * KOPT;                    // item_allocation [B,16,16]
    const size_t O2 = O1 + (size_t)B * 256;                // utility [16,B]
    const size_t O3 = O2 + (size_t)NAG * B;                // payments [16,B]
    const size_t O4 = O3 + (size_t)NAG * B;                // allocs_p [B,257,16,16]
    const size_t O5 = O4 + (size_t)B * KOPT * 256;         // bid_rev [16,B]

    __shared__ float vbw[256];          // w[n]*(val-bid), flat n*16+m
    __shared__ float paw[KOPT * 16];    // per-agent welfare [k][n]
    __shared__ float tw[KOPT];          // total welfare
    __shared__ float bsh[256];          // menu bias b
    __shared__ float ch[KOPT];          // alloc_choice
    __shared__ float ech[KOPT];         // exp scratch
    __shared__ float red[TPB];          // block reduction scratch
    __shared__ float cp[NAG];           // chosen per-agent welfare
    __shared__ float rcs[NAG];          // removed_chosen_sum
    __shared__ float rbs[NAG];          // removed_b
    __shared__ float ia[256];           // item_allocation

    const size_t bb = (size_t)b;
    const float* Ab = allocs + bb * MENU * 256;

    // ---- setup: vbw and b into LDS ----
    {
        int n = t >> 4;
        float bid = bids[bb * 256 + t];
        float val = vals[bb * 256 + t];
        vbw[t] = wv[bb * NAG + n] * (val - bid);
        bsh[t] = bvec_g[bb * 256 + t];
    }
    __syncthreads();

    // ---- Phase A: paw GEMM via WMMA f32 16x16x4, 2 k-tiles per wave ----
    for (int tile = 0; tile < 2; ++tile) {
        const int k0 = (wid * 2 + tile) * 16;
        const float* Arow = Ab + (size_t)(k0 + row) * 256;
        v8f acc = {};
#pragma unroll 8
        for (int s = 0; s < 64; ++s) {
            const int col = 4 * s + 2 * half;   // A layout: VGPR0<->K{0,2}, VGPR1<->K{1,3}
            float2 t2 = *reinterpret_cast<const float2*>(Arow + col); // global_load_b64
            v2f a; a.x = t2.x; a.y = t2.y;
            // block-diagonal B fragment: Bd[j,n] = (j>>4 == n) ? vbw[j] : 0
            const int hi = col >> 4;            // (col is even -> same for col and col+1)
            float bv0 = vbw[col];
            float bv1 = vbw[col + 1];
            v2f bf;
            bf.x = (hi == row) ? bv0 : 0.0f;
            bf.y = (hi == row) ? bv1 : 0.0f;
            acc = __builtin_amdgcn_wmma_f32_16x16x4_f32(
                false, a, false, bf, (short)0, acc, false, false);
        }
        // D layout: VGPR r, lanes 0-15 -> M=r, lanes 16-31 -> M=r+8; N=row
#pragma unroll
        for (int r = 0; r < 8; ++r)
            paw[(k0 + r + 8 * half) * 16 + row] = acc[r];
    }
    __syncthreads();

    // ---- Phase B: total welfare, softmax over K=257 ----
    {
        float s = 0.f;
#pragma unroll
        for (int n = 0; n < 16; ++n) s += paw[t * 16 + n];
        tw[t] = s;
        if (t == 0) tw[256] = 0.f;
        if (t < 16) paw[256 * 16 + t] = 0.f;   // null option welfare
    }
    __syncthreads();

    const float lg_t = (tw[t] + bsh[t]) * temp;  // logit for k=t; logit[256] == 0

    // max-reduce (thread 0 folds in the null option's logit 0)
    red[t] = (t == 0) ? fmaxf(lg_t, 0.0f) : lg_t;
    __syncthreads();
#pragma unroll
    for (int s = 128; s > 0; s >>= 1) {
        if (t < s) red[t] = fmaxf(red[t], red[t + s]);
        __syncthreads();
    }
    const float maxv = red[0];
    __syncthreads();

    const float e_t = expf(lg_t - maxv);
    ech[t] = e_t;
    float extra = 0.f;
    if (t == 0) { float e256 = expf(0.f - maxv); ech[256] = e256; extra = e256; }
    red[t] = e_t + extra;
    __syncthreads();
#pragma unroll
    for (int s = 128; s > 0; s >>= 1) {
        if (t < s) red[t] += red[t + s];
        __syncthreads();
    }
    const float inv_sum = 1.0f / red[0];
    __syncthreads();

    const float c_t = e_t * inv_sum;
    ch[t] = c_t;
    out[bb * KOPT + t] = c_t;                       // alloc_choice
    if (t == 0) {
        float c256 = ech[256] * inv_sum;
        ch[256] = c256;
        out[bb * KOPT + 256] = c256;
    }

    // alloc_b = sum_k choice[k]*b_p[k]  (null option contributes 0)
    red[t] = c_t * bsh[t];
    __syncthreads();
#pragma unroll
    for (int s = 128; s > 0; s >>= 1) {
        if (t < s) red[t] += red[t + s];
        __syncthreads();
    }
    const float allocb = red[0];
    __syncthreads();

    // chosen per-agent welfare
    if (t < NAG) {
        float s = 0.f;
        for (int k = 0; k < KOPT; ++k) s += ch[k] * paw[k * 16 + t];
        cp[t] = s;
    }

    // ---- Phase C: 16 counterfactual softmaxes, 2 agents per wave ----
    for (int ii = 0; ii < 2; ++ii) {
        const int i = wid * 2 + ii;
        float mx = -3.402823466e38f;
        for (int k = lane; k < KOPT; k += 32) {
            float tr = tw[k] - paw[k * 16 + i];
            float bp = (k < 256) ? bsh[k] : 0.f;
            mx = fmaxf(mx, (tr + bp) * temp);
        }
#pragma unroll
        for (int off = 16; off > 0; off >>= 1) mx = fmaxf(mx, __shfl_xor(mx, off, 32));
        float se = 0.f, str = 0.f, sb = 0.f;
        for (int k = lane; k < KOPT; k += 32) {
            float tr = tw[k] - paw[k * 16 + i];
            float bp = (k < 256) ? bsh[k] : 0.f;
            float e  = expf((tr + bp) * temp - mx);
            se += e; str += e * tr; sb += e * bp;
        }
#pragma unroll
        for (int off = 16; off > 0; off >>= 1) {
            se  += __shfl_xor(se,  off, 32);
            str += __shfl_xor(str, off, 32);
            sb  += __shfl_xor(sb,  off, 32);
        }
        if (lane == 0) { rcs[i] = str / se; rbs[i] = sb / se; }
    }

    // ---- Phase D: item_allocation + streamed allocs_p copy (coalesced) ----
    {
        float acc_ia = 0.f;
        float* outA = out + O4 + bb * KOPT * 256;
        for (int k = 0; k < MENU; ++k) {
            if ((k + 4) < MENU && (t & 31) == 0)   // 8 lanes/block -> 8 cachelines/row
                __builtin_prefetch(Ab + (size_t)(k + 4) * 256 + t, 0, 1);
            float a = Ab[(size_t)k * 256 + t];
            acc_ia = fmaf(ch[k], a, acc_ia);
            outA[(size_t)k * 256 + t] = a;
        }
        outA[(size_t)MENU * 256 + t] = 0.0f;       // null allocation row
        out[O1 + bb * 256 + t] = acc_ia;           // item_allocation
        ia[t] = acc_ia;
    }
    __syncthreads();

    // ---- Phase E: payments, utility, bid_rev ([n,B] layout) ----
    if (t < NAG) {
        float chosen_sum = 0.f;
#pragma unroll
        for (int n = 0; n < 16; ++n) chosen_sum += cp[n];
        float br = 0.f, vr = 0.f;
#pragma unroll
        for (int m = 0; m < 16; ++m) {
            float iam = ia[t * 16 + m];
            br = fmaf(iam, bids[bb * 256 + t * 16 + m], br);
            vr = fmaf(iam, vals[bb * 256 + t * 16 + m], vr);
        }
        float pay = (chosen_sum + allocb - rcs[t] - rbs[t]) / wv[bb * NAG + t] + br;
        out[O3 + (size_t)t * B + b] = pay;         // payments
        out[O2 + (size_t)t * B + b] = vr - pay;    // utility
        out[O5 + (size_t)t * B + b] = br;          // bid_rev
    }
}

extern "C" void kernel_launch(void* const* d_in, const int* in_sizes, int n_in,
                              void* d_out, int out_size, void* d_ws, size_t ws_size,
                              hipStream_t stream)
{
    const float* bids   = (const float*)d_in[0];
    const float* vals   = (const float*)d_in[1];
    const float* allocs = (const float*)d_in[2];
    const float* w      = (const float*)d_in[3];
    const float* bvec   = (const float*)d_in[4];
    const void*  temp   = d_in[5];
    const int B = in_sizes[3] / 16;   // w is [B,16]

    auction_menu_kernel<<<B, TPB, 0, stream>>>(bids, vals, allocs, w, bvec, temp,
                                               (float*)d_out, B);
}